// TransE_36352603193502
// MI455X (gfx1250) — compile-verified
//
#include <hip/hip_runtime.h>
#include <cstdint>

// TransE-with-word-pools scorer for MI455X (gfx1250, wave32).
//
// Geometry: 256-thread blocks = 8 wave32 waves; each wave processes 2 batch
// elements (16 lanes each). Lane ll in [0,16) covers dims [4*ll, 4*ll+4) of
// the 60-dim embedding (15 lanes * float4 = 60 exactly; lane 15 contributes
// zeros). Row L2-norms use a 4-step xor-butterfly within the 16-lane group.
//
// CDNA5-specific paths used:
//  * TENSOR_LOAD_TO_LDS (TDM, TENSORcnt) stages the per-block word-id tables.
//  * global_load_async_to_lds_b128 (ASYNCcnt) stages the per-block mask tables.
//  * s_wait_tensorcnt / s_wait_asynccnt before consuming staged data.
//  * wave32 ballots keep the 24-bit validity masks fully in registers.

constexpr int D       = 60;   // embedding dim
constexpr int L       = 24;   // words per list
constexpr int EPB     = 16;   // elements per block
constexpr int THREADS = 256;
constexpr int IDS_PER_BLK = EPB * L;   // 384 ints per table per block
#define EPSN 1e-12f

typedef unsigned int u32x4 __attribute__((ext_vector_type(4)));
typedef int          i32x8 __attribute__((ext_vector_type(8)));
typedef int          i32x4 __attribute__((ext_vector_type(4)));

__device__ __forceinline__ uint32_t lds_off(const void* p) {
  // Generic pointers to LDS carry the LDS byte offset in their low 32 bits.
  return (uint32_t)(uintptr_t)p;
}

__device__ __forceinline__ void async_load_lds_b128(uint32_t lds_addr, const void* gaddr) {
  asm volatile("global_load_async_to_lds_b128 %0, %1, off"
               :: "v"(lds_addr), "v"(gaddr) : "memory");
}

__device__ __forceinline__ void wait_async0() {
#if __has_builtin(__builtin_amdgcn_s_wait_asynccnt)
  __builtin_amdgcn_s_wait_asynccnt(0);
#else
  asm volatile("s_wait_asynccnt 0" ::: "memory");
#endif
}

__device__ __forceinline__ void wait_tensor0() {
#if __has_builtin(__builtin_amdgcn_s_wait_tensorcnt)
  __builtin_amdgcn_s_wait_tensorcnt(0);
#else
  asm volatile("s_wait_tensorcnt 0" ::: "memory");
#endif
}

// 1D TDM copy: copies nInts ints from global gaddr to the LDS byte address lds_addr, zero-fill past validInts.
__device__ __forceinline__ void tdm_copy_1d(uint32_t lds_addr, const void* gaddr,
                                            int nInts, int validInts) {
#if __has_builtin(__builtin_amdgcn_tensor_load_to_lds)
  const uint64_t ga = (uint64_t)(uintptr_t)gaddr;
  u32x4 g0;
  g0[0] = 0x1u;                                   // count=1, user descriptor
  g0[1] = lds_addr;                               // LDS byte address
  g0[2] = (uint32_t)ga;                           // global_addr[31:0]
  g0[3] = (uint32_t)((ga >> 32) & 0x01FFFFFFu)    // global_addr[56:32]
        | (2u << 30);                             // type=2 ("image")
  i32x8 g1;
  g1[0] = 0x20000;                                // data_size=2 (4B), no flags
  g1[1] = (validInts & 0xFFFF) << 16;             // tensor_dim0[15:0]
  g1[2] = ((validInts >> 16) & 0xFFFF)            // tensor_dim0[31:16]
        | (1 << 16);                              // tensor_dim1 = 1
  g1[3] = (nInts & 0xFFFF) << 16;                 // tile_dim0
  g1[4] = 0;                                      // tile_dim1/2 = 0 (1D tile)
  g1[5] = nInts;                                  // tensor_dim0_stride[31:0]
  g1[6] = 0;
  g1[7] = 0;
  i32x4 gz4 = {0, 0, 0, 0};
  i32x8 gz8 = {0, 0, 0, 0, 0, 0, 0, 0};
  // clang-23 / therock 6-arg form: (g0, g1, g2, g3, <extra v8i>, cpol)
  __builtin_amdgcn_tensor_load_to_lds(g0, g1, gz4, gz4, gz8, 0);
#else
  // Fallback: per-lane async copies (threads of the issuing wave).
  const int l = threadIdx.x & 31;
  for (int i = l * 4; i < nInts; i += 32 * 4)
    async_load_lds_b128(lds_addr + i * 4, (const int*)gaddr + i);
#endif
}

__device__ __forceinline__ uint32_t ballot32(bool pred) {
#if __has_builtin(__builtin_amdgcn_ballot_w32)
  return __builtin_amdgcn_ballot_w32(pred);
#else
  return (uint32_t)__ballot(pred ? 1 : 0);
#endif
}

__global__ __launch_bounds__(THREADS) void transe_score_kernel(
    const float* __restrict__ ent_emb, const float* __restrict__ rel_emb,
    const float* __restrict__ word_emb,
    const float* __restrict__ e_bias, const float* __restrict__ r_bias,
    const int* __restrict__ head, const int* __restrict__ relation,
    const int* __restrict__ tail,
    const int* __restrict__ head_w, const int* __restrict__ rel_w,
    const int* __restrict__ tail_w,
    const int* __restrict__ head_m, const int* __restrict__ rel_m,
    const int* __restrict__ tail_m,
    float* __restrict__ out, int nB)
{
  __shared__ __align__(16) int s_ids[3 * IDS_PER_BLK];   // [list][elem][word]
  __shared__ __align__(16) int s_msk[3 * IDS_PER_BLK];   // [list][elem][word]

  const int tid   = threadIdx.x;
  const int eBase = blockIdx.x * EPB;
  const int totalIds = nB * L;                    // ints per table
  const int blkBase  = eBase * L;
  const int remain   = totalIds - blkBase;        // >= 1 for launched blocks

  // ---- TDM-stage the three word-id tables for this block into LDS ----
  if ((tid >> 5) == 0) {                          // wave 0 issues the DMAs
    const int* srcs[3] = { head_w, rel_w, tail_w };
    #pragma unroll
    for (int j = 0; j < 3; ++j)
      tdm_copy_1d(lds_off(&s_ids[j * IDS_PER_BLK]), srcs[j] + blkBase,
                  IDS_PER_BLK, remain < IDS_PER_BLK ? remain : IDS_PER_BLK);
  }
  // ---- async-stage the three mask tables for this block into LDS ----
  {
    const int* msrc[3] = { head_m, rel_m, tail_m };
    if (tid < IDS_PER_BLK / 4) {                  // 96 threads x 16B
      int base = blkBase + tid * 4;
      if (base + 4 > totalIds) base = (totalIds >= 4) ? (totalIds - 4) : 0;
      #pragma unroll
      for (int j = 0; j < 3; ++j)
        async_load_lds_b128(lds_off(&s_msk[j * IDS_PER_BLK + tid * 4]),
                            msrc[j] + base);
    }
  }
  wait_tensor0();
  wait_async0();
  __syncthreads();

  const int lane = tid & 31;
  const int ll   = lane & 15;                 // lane within 16-lane group
  const int eLoc = (tid >> 5) * 2 + (lane >> 4);
  int e = eBase + eLoc;
  const bool valid = (e < nB);
  if (!valid) e = (nB > 0) ? (nB - 1) : 0;

  const int hid = head[e];
  const int rid = relation[e];
  const int tid2 = tail[e];

  float4 acc = make_float4(0.f, 0.f, 0.f, 0.f);

  // Normalize one gathered row and accumulate coef * row_hat into acc.
  auto accum_row = [&](const float* rowbase, float coef) {
    float4 v = make_float4(0.f, 0.f, 0.f, 0.f);
    if (ll < 15) v = *reinterpret_cast<const float4*>(rowbase + 4 * ll);
    float p = v.x * v.x;
    p = fmaf(v.y, v.y, p);
    p = fmaf(v.z, v.z, p);
    p = fmaf(v.w, v.w, p);
    p += __shfl_xor(p, 1);
    p += __shfl_xor(p, 2);
    p += __shfl_xor(p, 4);
    p += __shfl_xor(p, 8);
    float inv = coef * __builtin_amdgcn_rcpf(fmaxf(__builtin_amdgcn_sqrtf(p), EPSN));
    acc.x = fmaf(inv, v.x, acc.x);
    acc.y = fmaf(inv, v.y, acc.y);
    acc.z = fmaf(inv, v.z, acc.z);
    acc.w = fmaf(inv, v.w, acc.w);
  };

  // Base rows: normalize(ent[head]) + normalize(rel[r]) - normalize(ent[tail])
  accum_row(ent_emb + (size_t)hid  * D,  1.0f);
  accum_row(rel_emb + (size_t)rid  * D,  1.0f);
  accum_row(ent_emb + (size_t)tid2 * D, -1.0f);

  // Word pools: +head_pool +rel_pool -tail_pool
  #pragma unroll
  for (int list = 0; list < 3; ++list) {
    const int* mrow = &s_msk[list * IDS_PER_BLK + eLoc * L];
    int ma = mrow[ll];
    int mb = (ll < 8) ? mrow[16 + ll] : 0;
    uint32_t ba = ballot32(ma != 0);
    uint32_t bb = ballot32(mb != 0);
    // 24-bit validity mask for this element (bit l => word l valid)
    uint32_t m24 = (lane < 16)
        ? ((ba & 0xFFFFu) | ((bb & 0xFFu) << 16))
        : ((ba >> 16) | (((bb >> 16) & 0xFFu) << 16));
    // union over both halves -> wave-uniform skip mask (scalar branch)
    uint32_t uni = m24 | (uint32_t)__shfl_xor((int)m24, 16);
    uni = (uint32_t)__builtin_amdgcn_readfirstlane((int)uni);

    const float sign = (list == 2) ? -1.0f : 1.0f;
    const float cnt  = (float)__popc(m24);
    const float invc = sign * __builtin_amdgcn_rcpf(fmaxf(cnt, 1.0f));

    const int* ids = &s_ids[list * IDS_PER_BLK + eLoc * L];
    for (int l = 0; l < L; ++l) {
      if ((uni >> l) & 1u) {
        int wid = ids[l];                                   // ds_load, TDM-staged
        float mf = (float)((m24 >> l) & 1u);
        accum_row(word_emb + (size_t)wid * D, invc * mf);
      }
    }
  }

  // Final: score = -||acc|| + e_bias[h] + e_bias[t] + r_bias[r]
  float p = acc.x * acc.x;
  p = fmaf(acc.y, acc.y, p);
  p = fmaf(acc.z, acc.z, p);
  p = fmaf(acc.w, acc.w, p);
  p += __shfl_xor(p, 1);
  p += __shfl_xor(p, 2);
  p += __shfl_xor(p, 4);
  p += __shfl_xor(p, 8);

  float score = -__builtin_amdgcn_sqrtf(p)
              + e_bias[hid] + e_bias[tid2] + r_bias[rid];
  if (valid && ll == 0) out[e] = score;
}

extern "C" void kernel_launch(void* const* d_in, const int* in_sizes, int n_in,
                              void* d_out, int out_size, void* d_ws, size_t ws_size,
                              hipStream_t stream) {
  const float* ent_emb  = (const float*)d_in[0];
  const float* rel_emb  = (const float*)d_in[1];
  const float* word_emb = (const float*)d_in[2];
  const float* e_bias   = (const float*)d_in[3];
  const float* r_bias   = (const float*)d_in[4];
  const int*   head     = (const int*)d_in[5];
  const int*   relation = (const int*)d_in[6];
  const int*   tail     = (const int*)d_in[7];
  const int*   head_w   = (const int*)d_in[8];
  const int*   rel_w    = (const int*)d_in[9];
  const int*   tail_w   = (const int*)d_in[10];
  const int*   head_m   = (const int*)d_in[11];
  const int*   rel_m    = (const int*)d_in[12];
  const int*   tail_m   = (const int*)d_in[13];

  const int nB = in_sizes[5];
  if (nB <= 0) return;
  const int blocks = (nB + EPB - 1) / EPB;

  transe_score_kernel<<<blocks, THREADS, 0, stream>>>(
      ent_emb, rel_emb, word_emb, e_bias, r_bias,
      head, relation, tail,
      head_w, rel_w, tail_w,
      head_m, rel_m, tail_m,
      (float*)d_out, nB);
}